// GradientOrthogonalityRegularizer_19585050869843
// MI455X (gfx1250) — compile-verified
//
#include <hip/hip_runtime.h>
#include <hip/hip_bf16.h>
#include <math.h>

// ---- problem constants (match reference) ----
#define BB   2048   // batch
#define DD   512    // covariate dim
#define TT   16     // treatment dim
#define HH   1024   // mlp hidden
#define KIN  528    // D + T
#define KNB  50     // k neighbors
#define MM   16     // manifold dim
#define CSTR 516    // LDS row stride for C (512 + 4 pad)

#define ASTR 20     // A-tile LDS stride (16+4): 16 rows -> 16 distinct banks
#define BSTR 68     // B-tile LDS stride (64+4), 16B aligned rows

typedef float v2f __attribute__((ext_vector_type(2)));
typedef float v8f __attribute__((ext_vector_type(8)));
typedef int   vi4 __attribute__((ext_vector_type(4)));

// ---------------------------------------------------------------------------
// async global->LDS copy helpers (gfx1250 ASYNCcnt path with sync fallback)
// builtin signatures (from hipcc diagnostics): typed global src, LDS dst
// ---------------------------------------------------------------------------
#if __has_builtin(__builtin_amdgcn_global_load_async_to_lds_b128) && \
    __has_builtin(__builtin_amdgcn_global_load_async_to_lds_b32)
#define USE_ASYNC_LDS 1
#endif

__device__ __forceinline__ void async_cp16(void* lds, const void* g) {
#ifdef USE_ASYNC_LDS
  __builtin_amdgcn_global_load_async_to_lds_b128(
      (__attribute__((address_space(1))) vi4*)(g),
      (__attribute__((address_space(3))) vi4*)(lds), 0, 0);
#else
  *(float4*)lds = *(const float4*)g;
#endif
}

__device__ __forceinline__ void async_cp4(void* lds, const void* g) {
#ifdef USE_ASYNC_LDS
  __builtin_amdgcn_global_load_async_to_lds_b32(
      (__attribute__((address_space(1))) int*)(g),
      (__attribute__((address_space(3))) int*)(lds), 0, 0);
#else
  *(float*)lds = *(const float*)g;
#endif
}

__device__ __forceinline__ void async_wait() {
#ifdef USE_ASYNC_LDS
#if __has_builtin(__builtin_amdgcn_s_wait_asynccnt)
  __builtin_amdgcn_s_wait_asynccnt(0);
#else
  asm volatile("s_wait_asynccnt 0x0" ::: "memory");
#endif
#endif
}

// ---------------------------------------------------------------------------
// 1) row squared norms of perturbations
// ---------------------------------------------------------------------------
__global__ __launch_bounds__(256) void k_sq(const float* __restrict__ P,
                                            float* __restrict__ sq) {
  __shared__ float s[256];
  const int row = blockIdx.x, tid = threadIdx.x;
  float a = 0.f;
  for (int d = tid; d < DD; d += 256) { float v = P[row * DD + d]; a += v * v; }
  s[tid] = a; __syncthreads();
  for (int st = 128; st > 0; st >>= 1) {
    if (tid < st) s[tid] += s[tid + st];
    __syncthreads();
  }
  if (tid == 0) sq[row] = s[0];
}

// ---------------------------------------------------------------------------
// 2) forward GEMM  a = [x|t] @ W1 + b1 ; fused g = (1-tanh(a)^2)*W2
//    block: 256 thr / 8 waves, 64x64 tile, K staged 16-wide in LDS (async)
// ---------------------------------------------------------------------------
__global__ __launch_bounds__(256) void k_gemm_act(const float* __restrict__ x,
                                                  const float* __restrict__ t,
                                                  const float* __restrict__ W1,
                                                  const float* __restrict__ b1,
                                                  const float* __restrict__ W2,
                                                  float* __restrict__ gbuf) {
  __shared__ float As[64 * ASTR];
  __shared__ float Bs[16 * BSTR];
  const int tid = threadIdx.x;
  const int rowBase = blockIdx.y * 64;
  const int colBase = blockIdx.x * 64;
  const int wave = tid >> 5, lane = tid & 31;
  const int mr = lane & 15, ks = (lane >> 4) * 2, half = lane >> 4;
  const int tm0 = wave >> 2, tn = wave & 3;       // wave tiles: (tm0,tn),(tm0+2,tn)
  const int ar = tid >> 2, akc = (tid & 3) * 4;   // A staging: row, k-offset
  const int bk = tid >> 4, bnc = (tid & 15) * 4;  // B staging: k-row, n-offset
  v8f acc0 = {}, acc1 = {};
  for (int k0 = 0; k0 < KIN; k0 += 16) {
    __syncthreads();
    if (k0 < DD)
      async_cp16(&As[ar * ASTR + akc], &x[(size_t)(rowBase + ar) * DD + k0 + akc]);
    else
      async_cp16(&As[ar * ASTR + akc], &t[(size_t)(rowBase + ar) * TT + (k0 - DD) + akc]);
    async_cp16(&Bs[bk * BSTR + bnc], &W1[(size_t)(k0 + bk) * HH + colBase + bnc]);
    async_wait();
    __syncthreads();
#pragma unroll
    for (int kk = 0; kk < 16; kk += 4) {
      const int ka = kk + ks;
      v2f b;
      b.x = Bs[(ka    ) * BSTR + tn * 16 + mr];
      b.y = Bs[(ka + 1) * BSTR + tn * 16 + mr];
      const v2f a0 = *(const v2f*)&As[((tm0    ) * 16 + mr) * ASTR + ka];
      const v2f a1 = *(const v2f*)&As[((tm0 + 2) * 16 + mr) * ASTR + ka];
      acc0 = __builtin_amdgcn_wmma_f32_16x16x4_f32(false, a0, false, b,
                                                   (short)0, acc0, false, false);
      acc1 = __builtin_amdgcn_wmma_f32_16x16x4_f32(false, a1, false, b,
                                                   (short)0, acc1, false, false);
    }
  }
  const int c = colBase + tn * 16 + mr;
  const float w2c = W2[c], b1c = b1[c];
#pragma unroll
  for (int v = 0; v < 8; ++v) {
    const int r0 = rowBase + tm0 * 16 + v + half * 8;
    const float th0 = tanhf(acc0[v] + b1c);
    gbuf[(size_t)r0 * HH + c] = (1.f - th0 * th0) * w2c;
    const float th1 = tanhf(acc1[v] + b1c);
    gbuf[(size_t)(r0 + 32) * HH + c] = (1.f - th1 * th1) * w2c;
  }
}

// ---------------------------------------------------------------------------
// 3) grad_x = g @ W1[:D,:]^T   (M=2048, N=512, K=1024); B is transposed in mem
// ---------------------------------------------------------------------------
__global__ __launch_bounds__(256) void k_gemm_grad(const float* __restrict__ gbuf,
                                                   const float* __restrict__ W1,
                                                   float* __restrict__ gradx) {
  __shared__ float As[64 * ASTR];
  __shared__ float Bs[16 * BSTR];
  const int tid = threadIdx.x;
  const int rowBase = blockIdx.y * 64;
  const int colBase = blockIdx.x * 64;
  const int wave = tid >> 5, lane = tid & 31;
  const int mr = lane & 15, ks = (lane >> 4) * 2, half = lane >> 4;
  const int tm0 = wave >> 2, tn = wave & 3;
  const int ar = tid >> 2, akc = (tid & 3) * 4;
  const int bn = tid >> 2, bkc = (tid & 3) * 4;   // B staging: n-row, k-offset
  v8f acc0 = {}, acc1 = {};
  for (int k0 = 0; k0 < HH; k0 += 16) {
    __syncthreads();
    async_cp16(&As[ar * ASTR + akc], &gbuf[(size_t)(rowBase + ar) * HH + k0 + akc]);
    {  // W1^T[k][n] = W1[n*H + k]: load 4 along k, scatter-transpose into Bs
      const float* src = &W1[(size_t)(colBase + bn) * HH + k0 + bkc];
      async_cp4(&Bs[(bkc + 0) * BSTR + bn], src + 0);
      async_cp4(&Bs[(bkc + 1) * BSTR + bn], src + 1);
      async_cp4(&Bs[(bkc + 2) * BSTR + bn], src + 2);
      async_cp4(&Bs[(bkc + 3) * BSTR + bn], src + 3);
    }
    async_wait();
    __syncthreads();
#pragma unroll
    for (int kk = 0; kk < 16; kk += 4) {
      const int ka = kk + ks;
      v2f b;
      b.x = Bs[(ka    ) * BSTR + tn * 16 + mr];
      b.y = Bs[(ka + 1) * BSTR + tn * 16 + mr];
      const v2f a0 = *(const v2f*)&As[((tm0    ) * 16 + mr) * ASTR + ka];
      const v2f a1 = *(const v2f*)&As[((tm0 + 2) * 16 + mr) * ASTR + ka];
      acc0 = __builtin_amdgcn_wmma_f32_16x16x4_f32(false, a0, false, b,
                                                   (short)0, acc0, false, false);
      acc1 = __builtin_amdgcn_wmma_f32_16x16x4_f32(false, a1, false, b,
                                                   (short)0, acc1, false, false);
    }
  }
  const int c = colBase + tn * 16 + mr;
#pragma unroll
  for (int v = 0; v < 8; ++v) {
    const int r0 = rowBase + tm0 * 16 + v + half * 8;
    gradx[(size_t)r0 * DD + c] = acc0[v];
    gradx[(size_t)(r0 + 32) * DD + c] = acc1[v];
  }
}

// ---------------------------------------------------------------------------
// 4) pairwise squared distances  d2 = sq_i + sq_j - 2 P P^T
// ---------------------------------------------------------------------------
__global__ __launch_bounds__(256) void k_gemm_dist(const float* __restrict__ P,
                                                   const float* __restrict__ sq,
                                                   float* __restrict__ d2) {
  __shared__ float As[64 * ASTR];
  __shared__ float Bs[16 * BSTR];
  const int tid = threadIdx.x;
  const int rowBase = blockIdx.y * 64;
  const int colBase = blockIdx.x * 64;
  const int wave = tid >> 5, lane = tid & 31;
  const int mr = lane & 15, ks = (lane >> 4) * 2, half = lane >> 4;
  const int tm0 = wave >> 2, tn = wave & 3;
  const int ar = tid >> 2, akc = (tid & 3) * 4;
  const int bn = tid >> 2, bkc = (tid & 3) * 4;
  v8f acc0 = {}, acc1 = {};
  for (int k0 = 0; k0 < DD; k0 += 16) {
    __syncthreads();
    async_cp16(&As[ar * ASTR + akc], &P[(size_t)(rowBase + ar) * DD + k0 + akc]);
    {  // P^T[k][n] = P[n*D + k]
      const float* src = &P[(size_t)(colBase + bn) * DD + k0 + bkc];
      async_cp4(&Bs[(bkc + 0) * BSTR + bn], src + 0);
      async_cp4(&Bs[(bkc + 1) * BSTR + bn], src + 1);
      async_cp4(&Bs[(bkc + 2) * BSTR + bn], src + 2);
      async_cp4(&Bs[(bkc + 3) * BSTR + bn], src + 3);
    }
    async_wait();
    __syncthreads();
#pragma unroll
    for (int kk = 0; kk < 16; kk += 4) {
      const int ka = kk + ks;
      v2f b;
      b.x = Bs[(ka    ) * BSTR + tn * 16 + mr];
      b.y = Bs[(ka + 1) * BSTR + tn * 16 + mr];
      const v2f a0 = *(const v2f*)&As[((tm0    ) * 16 + mr) * ASTR + ka];
      const v2f a1 = *(const v2f*)&As[((tm0 + 2) * 16 + mr) * ASTR + ka];
      acc0 = __builtin_amdgcn_wmma_f32_16x16x4_f32(false, a0, false, b,
                                                   (short)0, acc0, false, false);
      acc1 = __builtin_amdgcn_wmma_f32_16x16x4_f32(false, a1, false, b,
                                                   (short)0, acc1, false, false);
    }
  }
  const int c = colBase + tn * 16 + mr;
  const float sqc = sq[c];
#pragma unroll
  for (int v = 0; v < 8; ++v) {
    const int r0 = rowBase + tm0 * 16 + v + half * 8;
    d2[(size_t)r0 * BB + c] = sq[r0] + sqc - 2.f * acc0[v];
    d2[(size_t)(r0 + 32) * BB + c] = sq[r0 + 32] + sqc - 2.f * acc1[v];
  }
}

// ---------------------------------------------------------------------------
// 5) top-K (smallest) per row via iterated LDS argmin (deterministic tiebreak)
// ---------------------------------------------------------------------------
__global__ __launch_bounds__(256) void k_topk(const float* __restrict__ d2,
                                              int* __restrict__ idxb) {
  __shared__ float sd[BB];
  __shared__ float rv[256];
  __shared__ int   ri[256];
  const int row = blockIdx.x, tid = threadIdx.x;
  for (int j = tid; j < BB; j += 256) sd[j] = d2[(size_t)row * BB + j];
  __syncthreads();
  for (int it = 0; it < KNB; ++it) {
    float best = 3.0e38f; int bi = 0x7fffffff;
    for (int j = tid; j < BB; j += 256) {
      const float v = sd[j];
      if (v < best || (v == best && j < bi)) { best = v; bi = j; }
    }
    rv[tid] = best; ri[tid] = bi;
    __syncthreads();
    for (int st = 128; st > 0; st >>= 1) {
      if (tid < st) {
        const float vo = rv[tid + st]; const int io = ri[tid + st];
        if (vo < rv[tid] || (vo == rv[tid] && io < ri[tid])) { rv[tid] = vo; ri[tid] = io; }
      }
      __syncthreads();
    }
    if (tid == 0) { idxb[row * KNB + it] = ri[0]; sd[ri[0]] = 3.0e38f; }
    __syncthreads();
  }
}

// ---------------------------------------------------------------------------
// 6) per-row tangent-space projection energy:
//    async gather -> center -> G = C C^T (fp32 WMMA from LDS) -> subspace
//    iteration -> value = y^T H^{-1} y  (H = Q^T G Q, y = Q^T C g)
// ---------------------------------------------------------------------------
__global__ __launch_bounds__(256) void k_tangent(const float* __restrict__ P,
                                                 const int* __restrict__ idxb,
                                                 const float* __restrict__ gradx,
                                                 float* __restrict__ rowval) {
  extern __shared__ float sm[];
  float* Cs = sm;                    // KNB * CSTR
  float* gv = Cs + KNB * CSTR;       // DD
  float* w  = gv + DD;               // 64
  float* G  = w + 64;                // 64 * 66
  float* Q  = G + 64 * 66;           // 64 * 17
  float* Z  = Q + 64 * 17;           // 64 * 17
  float* rr = Z + 64 * 17;           // 64
  float* Hm = rr + 64;               // 256
  float* yv = Hm + 256;              // 16
  float* zv = yv + 16;               // 16
  __shared__ int nidx[64];

  const int row = blockIdx.x;
  const int tid = threadIdx.x;

  if (tid < KNB) nidx[tid] = idxb[row * KNB + tid];
  __syncthreads();

  // async gather of neighbor rows (50 x 512 floats) + gradient row
  for (int e = tid; e < KNB * (DD / 4); e += 256) {
    const int kk = e / (DD / 4);
    const int d4 = (e % (DD / 4)) * 4;
    async_cp16(&Cs[kk * CSTR + d4], &P[(size_t)nidx[kk] * DD + d4]);
  }
  for (int d = tid; d < DD; d += 256) gv[d] = gradx[(size_t)row * DD + d];
  async_wait();
  __syncthreads();

  // center columns over the K neighbors
  for (int d = tid; d < DD; d += 256) {
    float mu = 0.f;
    for (int kk = 0; kk < KNB; ++kk) mu += Cs[kk * CSTR + d];
    mu *= (1.f / KNB);
    for (int kk = 0; kk < KNB; ++kk) Cs[kk * CSTR + d] -= mu;
  }
  __syncthreads();

  // w = C g
  if (tid < 64) {
    float s = 0.f;
    if (tid < KNB)
      for (int d = 0; d < DD; ++d) s += Cs[tid * CSTR + d] * gv[d];
    w[tid] = s;
  }
  __syncthreads();

  // G = C C^T via fp32 WMMA (64x64 padded; rows >= KNB are zero)
  {
    const int wave = tid >> 5, lane = tid & 31;
    const int mr = lane & 15, ks = (lane >> 4) * 2, nh = lane >> 4;
    for (int tt = wave; tt < 16; tt += 8) {
      const int tm = tt >> 2, tn = tt & 3;
      const int ra = tm * 16 + mr;
      const int rb = tn * 16 + mr;
      v8f acc = {};
      for (int k0 = 0; k0 < DD; k0 += 4) {
        const int ka = k0 + ks;
        v2f a, b;
        a.x = (ra < KNB) ? Cs[ra * CSTR + ka]     : 0.f;
        a.y = (ra < KNB) ? Cs[ra * CSTR + ka + 1] : 0.f;
        b.x = (rb < KNB) ? Cs[rb * CSTR + ka]     : 0.f;
        b.y = (rb < KNB) ? Cs[rb * CSTR + ka + 1] : 0.f;
        acc = __builtin_amdgcn_wmma_f32_16x16x4_f32(false, a, false, b,
                                                    (short)0, acc, false, false);
      }
      for (int v = 0; v < 8; ++v)
        G[(tm * 16 + v + nh * 8) * 66 + tn * 16 + mr] = acc[v];
    }
  }
  __syncthreads();

  // deterministic pseudo-random init of Q (50 x 16, padded to 64 rows)
  for (int e = tid; e < 64 * MM; e += 256) {
    const int kk = e >> 4, j = e & 15;
    unsigned h = (unsigned)(kk * 73856093u) ^ (unsigned)(j * 19349663u) ^ 0x9e3779b9u;
    h ^= h >> 13; h *= 0x85ebca6bu; h ^= h >> 16;
    Q[kk * 17 + j] = (kk < KNB) ? ((float)(h & 0xFFFFu) / 32768.f - 1.f) : 0.f;
  }
  __syncthreads();

  // block subspace iteration: 8x (Z = G Q ; MGS -> Q), then final Z = G Q
  for (int it = 0; it < 9; ++it) {
    for (int e = tid; e < KNB * MM; e += 256) {
      const int kk = e / MM, j = e % MM;
      float s = 0.f;
      for (int l = 0; l < KNB; ++l) s += G[kk * 66 + l] * Q[l * 17 + j];
      Z[kk * 17 + j] = s;
    }
    __syncthreads();
    if (it == 8) break;
    for (int j = 0; j < MM; ++j) {
      if (tid < j) {
        float s = 0.f;
        for (int kk = 0; kk < KNB; ++kk) s += Q[kk * 17 + tid] * Z[kk * 17 + j];
        rr[tid] = s;
      }
      __syncthreads();
      if (tid < KNB) {
        float v = Z[tid * 17 + j];
        for (int i = 0; i < j; ++i) v -= rr[i] * Q[tid * 17 + i];
        Z[tid * 17 + j] = v;
      }
      __syncthreads();
      if (tid == 0) {
        float s = 0.f;
        for (int kk = 0; kk < KNB; ++kk) { const float v = Z[kk * 17 + j]; s += v * v; }
        rr[63] = 1.f / sqrtf(s + 1e-30f);
      }
      __syncthreads();
      if (tid < 64) Q[tid * 17 + j] = (tid < KNB) ? Z[tid * 17 + j] * rr[63] : 0.f;
      __syncthreads();
    }
  }

  // H = Q^T Z (= Q^T G Q), y = Q^T w
  {
    const int i = tid >> 4, j = tid & 15;
    float s = 0.f;
    for (int kk = 0; kk < KNB; ++kk) s += Q[kk * 17 + i] * Z[kk * 17 + j];
    Hm[i * 16 + j] = s;
  }
  if (tid < MM) {
    float s = 0.f;
    for (int kk = 0; kk < KNB; ++kk) s += Q[kk * 17 + tid] * w[kk];
    yv[tid] = s;
  }
  __syncthreads();

  if (tid == 0) {
    float tr = 0.f;
    for (int i = 0; i < MM; ++i) tr += Hm[i * 16 + i];
    const float eps = 1e-7f * tr + 1e-20f;
    for (int i = 0; i < MM; ++i) Hm[i * 16 + i] += eps;
    // in-place lower Cholesky
    for (int i = 0; i < MM; ++i)
      for (int j = 0; j <= i; ++j) {
        float s = Hm[i * 16 + j];
        for (int p = 0; p < j; ++p) s -= Hm[i * 16 + p] * Hm[j * 16 + p];
        if (i == j) Hm[i * 16 + i] = sqrtf(fmaxf(s, 1e-30f));
        else        Hm[i * 16 + j] = s / Hm[j * 16 + j];
      }
    // solve L u = y ;  y^T H^{-1} y = ||u||^2
    float val = 0.f;
    for (int i = 0; i < MM; ++i) {
      float s = yv[i];
      for (int p = 0; p < i; ++p) s -= Hm[i * 16 + p] * zv[p];
      zv[i] = s / Hm[i * 16 + i];
    }
    for (int i = 0; i < MM; ++i) val += zv[i] * zv[i];
    rowval[row] = val;
  }
}

// ---------------------------------------------------------------------------
// 7) deterministic final reduction (no float atomics -> replay-stable)
// ---------------------------------------------------------------------------
__global__ __launch_bounds__(256) void k_reduce(const float* __restrict__ rowval,
                                                float* __restrict__ out) {
  __shared__ float s[256];
  float acc = 0.f;
  for (int i = threadIdx.x; i < BB; i += 256) acc += rowval[i];
  s[threadIdx.x] = acc; __syncthreads();
  for (int st = 128; st > 0; st >>= 1) {
    if (threadIdx.x < st) s[threadIdx.x] += s[threadIdx.x + st];
    __syncthreads();
  }
  if (threadIdx.x == 0) out[0] = s[0] / (float)BB;
}

// ---------------------------------------------------------------------------
extern "C" void kernel_launch(void* const* d_in, const int* in_sizes, int n_in,
                              void* d_out, int out_size, void* d_ws, size_t ws_size,
                              hipStream_t stream) {
  (void)in_sizes; (void)n_in; (void)out_size; (void)ws_size;
  const float* x  = (const float*)d_in[0];
  const float* t  = (const float*)d_in[1];
  const float* P  = (const float*)d_in[2];   // perturbations
  const float* W1 = (const float*)d_in[3];
  const float* b1 = (const float*)d_in[4];
  const float* W2 = (const float*)d_in[5];
  // d_in[6] = b2: constant offset, vanishes under d/dx -> unused
  float* out = (float*)d_out;

  float* gbuf   = (float*)d_ws;              // BB*HH
  float* gradx  = gbuf  + (size_t)BB * HH;   // BB*DD
  float* d2     = gradx + (size_t)BB * DD;   // BB*BB
  float* sq     = d2    + (size_t)BB * BB;   // BB
  int*   idxb   = (int*)(sq + BB);           // BB*KNB
  float* rowval = (float*)(idxb + (size_t)BB * KNB); // BB

  k_sq       <<<BB, 256, 0, stream>>>(P, sq);
  k_gemm_act <<<dim3(HH / 64, BB / 64), 256, 0, stream>>>(x, t, W1, b1, W2, gbuf);
  k_gemm_grad<<<dim3(DD / 64, BB / 64), 256, 0, stream>>>(gbuf, W1, gradx);
  k_gemm_dist<<<dim3(BB / 64, BB / 64), 256, 0, stream>>>(P, sq, d2);
  k_topk     <<<BB, 256, 0, stream>>>(d2, idxb);

  const size_t smem = (size_t)(KNB * CSTR + DD + 64 + 64 * 66 + 64 * 17 + 64 * 17 +
                               64 + 256 + 16 + 16) * sizeof(float);
  k_tangent  <<<BB, 256, smem, stream>>>(P, idxb, gradx, rowval);
  k_reduce   <<<1, 256, 0, stream>>>(rowval, out);
}